// ImprovedGATv2_53463752900653
// MI455X (gfx1250) — compile-verified
//
#include <hip/hip_runtime.h>
#include <cstdint>
#include <cstddef>

typedef float v2f __attribute__((ext_vector_type(2)));
typedef float v8f __attribute__((ext_vector_type(8)));

#define NEG_SLOPE 0.2f
#define LN_EPS 1e-5f
#define FLT_MAX_NEG (-3.402823466e38f)

// ---------------------------------------------------------------- fill
__global__ void k_fill(float* __restrict__ p, float v, size_t n) {
  size_t i = (size_t)blockIdx.x * blockDim.x + threadIdx.x;
  if (i < n) p[i] = v;
}

// ------------------------------------------- GEMM via f32 WMMA 16x16x4
// Y[nrows, NC] = X[nrows, K] @ W[K, NC] + bias.  One wave per 16-row strip,
// NC/16 accumulator tiles per wave. nrows % 16 == 0 (50000 = 3125*16).
// A-frag (16x4 f32): lane L<16 -> M=L, VGPR0=K0, VGPR1=K1; lane>=16 -> K2,K3.
// B-frag (4x16 f32): VGPR0: lanes0-15 row K0 / lanes16-31 row K2; VGPR1: K1/K3.
// C/D (16x16 f32): VGPR v: lanes0-15 -> M=v, lanes16-31 -> M=v+8, N=lane&15.
template<int K, int NC>
__global__ void k_gemm_bias(const float* __restrict__ X,
                            const float* __restrict__ Wm,
                            const float* __restrict__ bias,
                            float* __restrict__ Y, int nrows) {
  const int wid  = (int)((blockIdx.x * blockDim.x + threadIdx.x) >> 5);
  const int lane = threadIdx.x & 31;
  const int row0 = wid * 16;
  if (row0 >= nrows) return;              // wave-uniform
  constexpr int NT = NC / 16;
  v8f c[NT];
#pragma unroll
  for (int t = 0; t < NT; ++t)
#pragma unroll
    for (int q = 0; q < 8; ++q) c[t][q] = 0.f;

  const int mA   = lane & 15;
  const int kOff = (lane >> 4) << 1;      // 0 or 2
  const float* xrow = X + (size_t)(row0 + mA) * K;

  for (int k = 0; k < K; k += 4) {
    v2f a;
    a[0] = xrow[k + kOff + 0];
    a[1] = xrow[k + kOff + 1];
#pragma unroll
    for (int t = 0; t < NT; ++t) {
      const float* wp = Wm + (size_t)(k + kOff) * NC + t * 16 + mA;
      v2f b;
      b[0] = wp[0];
      b[1] = wp[NC];
      c[t] = __builtin_amdgcn_wmma_f32_16x16x4_f32(
          false, a, false, b, (short)0, c[t], false, false);
    }
  }
  const int rOff = (lane >> 4) << 3;      // 0 or 8
  const int col  = lane & 15;
#pragma unroll
  for (int t = 0; t < NT; ++t) {
    float bv = bias[t * 16 + col];
#pragma unroll
    for (int v = 0; v < 8; ++v) {
      int r = row0 + rOff + v;
      Y[(size_t)r * NC + t * 16 + col] = c[t][v] + bv;
    }
  }
}

// ------------------------------------------------------- float atomic max
__device__ inline void atomicMaxF(float* addr, float v) {
  if (v >= 0.f) atomicMax((int*)addr, __float_as_int(v));
  else          atomicMin((unsigned int*)addr, __float_as_uint(v));
}

// -------------------------------------------- edge scores + segment max
// wave per edge; lane l handles elements i = l + 32*j of the H*C vector.
template<int H, int C>
__global__ void k_edge_scores(const float* __restrict__ xl,
                              const float* __restrict__ xr,
                              const int* __restrict__ ei, int nraw, int ntot,
                              const float* __restrict__ att,
                              float* __restrict__ escore,
                              float* __restrict__ emax) {
  const int e    = (int)((blockIdx.x * blockDim.x + threadIdx.x) >> 5);
  const int lane = threadIdx.x & 31;
  if (e >= ntot) return;                  // wave-uniform
  int s, d;
  if (e < nraw) { s = ei[e]; d = ei[nraw + e]; }
  else          { s = d = e - nraw; }
  constexpr int HC = H * C;
  constexpr int J  = HC / 32;
  float r[J];
#pragma unroll
  for (int j = 0; j < J; ++j) {
    int i   = lane + 32 * j;
    float m = xl[(size_t)s * HC + i] + xr[(size_t)d * HC + i];
    m       = (m > 0.f) ? m : NEG_SLOPE * m;
    r[j]    = m * att[i];
  }
  if constexpr (C == 32) {                // each j is one head
#pragma unroll
    for (int j = 0; j < J; ++j)
#pragma unroll
      for (int msk = 16; msk >= 1; msk >>= 1) r[j] += __shfl_xor(r[j], msk, 32);
    float my = r[0];
#pragma unroll
    for (int h = 1; h < H; ++h) if (lane == h) my = r[h];
    if (lane < H) {
      escore[(size_t)e * H + lane] = my;
      atomicMaxF(&emax[(size_t)d * H + lane], my);
    }
  } else {                                // single head spanning all lanes
    float t = 0.f;
#pragma unroll
    for (int j = 0; j < J; ++j) t += r[j];
#pragma unroll
    for (int msk = 16; msk >= 1; msk >>= 1) t += __shfl_xor(t, msk, 32);
    if (lane == 0) {
      escore[e] = t;
      atomicMaxF(&emax[d], t);
    }
  }
}

// -------------------------------------------------- exp + segment denom
template<int H>
__global__ void k_edge_exp(const int* __restrict__ ei, int nraw, int ntot,
                           float* __restrict__ escore,
                           const float* __restrict__ emax,
                           float* __restrict__ denom) {
  int idx = blockIdx.x * blockDim.x + threadIdx.x;
  if (idx >= ntot * H) return;
  int e = idx / H;
  int h = idx - e * H;
  int d = (e < nraw) ? ei[nraw + e] : e - nraw;
  float ex = expf(escore[idx] - emax[(size_t)d * H + h]);
  escore[idx] = ex;                       // escore now holds exp values
  atomicAdd(&denom[(size_t)d * H + h], ex);
}

// ------------------------------------------- weighted scatter (wave/edge)
template<int H, int C>
__global__ void k_edge_scatter(const float* __restrict__ xl,
                               const int* __restrict__ ei, int nraw, int ntot,
                               const float* __restrict__ exs,
                               const float* __restrict__ denom,
                               float* __restrict__ acc) {
  const int e    = (int)((blockIdx.x * blockDim.x + threadIdx.x) >> 5);
  const int lane = threadIdx.x & 31;
  if (e >= ntot) return;
  int s, d;
  if (e < nraw) { s = ei[e]; d = ei[nraw + e]; }
  else          { s = d = e - nraw; }
  constexpr int HC = H * C;
  float alpha[H];
#pragma unroll
  for (int h = 0; h < H; ++h)
    alpha[h] = exs[(size_t)e * H + h] / (denom[(size_t)d * H + h] + 1e-16f);
#pragma unroll
  for (int j = 0; j < HC / 32; ++j) {
    int i = lane + 32 * j;
    atomicAdd(&acc[(size_t)d * HC + i],
              xl[(size_t)s * HC + i] * alpha[(32 * j) / C]);
  }
}

// ------------------------------- node: +bias +res, LayerNorm(128), ELU
__global__ void k_node_finish(const float* __restrict__ acc,
                              const float* __restrict__ res,
                              const float* __restrict__ bias,
                              const float* __restrict__ gamma,
                              const float* __restrict__ beta,
                              float* __restrict__ out, int n) {
  const int node = (int)((blockIdx.x * blockDim.x + threadIdx.x) >> 5);
  const int lane = threadIdx.x & 31;
  if (node >= n) return;
  float v[4];
  float sum = 0.f;
#pragma unroll
  for (int j = 0; j < 4; ++j) {
    int i = lane + 32 * j;
    v[j]  = acc[(size_t)node * 128 + i] + bias[i] + res[(size_t)node * 128 + i];
    sum  += v[j];
  }
#pragma unroll
  for (int m = 16; m >= 1; m >>= 1) sum += __shfl_xor(sum, m, 32);
  float mu  = sum * (1.f / 128.f);
  float var = 0.f;
#pragma unroll
  for (int j = 0; j < 4; ++j) { float t = v[j] - mu; var += t * t; }
#pragma unroll
  for (int m = 16; m >= 1; m >>= 1) var += __shfl_xor(var, m, 32);
  var *= (1.f / 128.f);
  float rs = rsqrtf(var + LN_EPS);
#pragma unroll
  for (int j = 0; j < 4; ++j) {
    int i   = lane + 32 * j;
    float y = (v[j] - mu) * rs * gamma[i] + beta[i];
    out[(size_t)node * 128 + i] = (y > 0.f) ? y : (expf(y) - 1.f);  // ELU
  }
}

// --------------------------------------------------- layer-2 final bias
__global__ void k_final_bias(const float* __restrict__ acc,
                             const float* __restrict__ bias,
                             float* __restrict__ out, int n) {
  int idx = blockIdx.x * blockDim.x + threadIdx.x;
  if (idx >= n * 64) return;
  out[idx] = acc[idx] + bias[idx & 63];
}

// =====================================================================
extern "C" void kernel_launch(void* const* d_in, const int* in_sizes, int n_in,
                              void* d_out, int out_size, void* d_ws, size_t ws_size,
                              hipStream_t stream) {
  constexpr int N  = 50000;
  constexpr int E  = 800000;
  constexpr int ET = E + N;               // self-loops appended

  auto f = [&](int i) { return (const float*)d_in[i]; };

  // ---- input mapping: handle insertion-order, sorted-inner, full-sorted ----
  const float* xin;
  const int*   ei;
  int  pbase = 1;
  bool innerSorted;
  if (in_sizes[0] == 2 * E) {             // fully-sorted pytree: edge_index first
    ei = (const int*)d_in[0];
    xin = f(27);
    innerSorted = true;
  } else {                                // top-level insertion order
    xin = f(0);
    ei = (const int*)d_in[27];
    innerSorted = (in_sizes[1] != 256 * 128);  // wl first => insertion order
  }
  const float *WL[3], *BL[3], *WR[3], *BR[3], *ATT[3], *BIAS[3];
  const float *GAM[2], *BET[2], *RW[2], *RB[2];
  for (int l = 0; l < 3; ++l) {
    int p = pbase + 6 * l;
    if (innerSorted) { ATT[l]=f(p); BIAS[l]=f(p+1); BL[l]=f(p+2); BR[l]=f(p+3); WL[l]=f(p+4); WR[l]=f(p+5); }
    else             { WL[l]=f(p);  BL[l]=f(p+1);  WR[l]=f(p+2); BR[l]=f(p+3); ATT[l]=f(p+4); BIAS[l]=f(p+5); }
  }
  for (int l = 0; l < 2; ++l) {
    int p = pbase + 18 + 2 * l;
    if (innerSorted) { BET[l]=f(p); GAM[l]=f(p+1); }
    else             { GAM[l]=f(p); BET[l]=f(p+1); }
  }
  for (int l = 0; l < 2; ++l) {
    int p = pbase + 22 + 2 * l;
    if (innerSorted) { RB[l]=f(p); RW[l]=f(p+1); }
    else             { RW[l]=f(p); RB[l]=f(p+1); }
  }

  // ---- workspace layout (floats): fits in L2 (~118 MB total) ----
  float* ws   = (float*)d_ws;
  float* XL   = ws;                        // x_l               [N,128]
  float* XRA  = XL  + (size_t)N * 128;     // x_r then acc      [N,128]
  float* RES  = XRA + (size_t)N * 128;     // residual          [N,128]
  float* F0   = RES + (size_t)N * 128;     // features ping     [N,128]
  float* ESC  = F0  + (size_t)N * 128;     // edge scores/exp   [ET,4]
  float* EMAX = ESC + (size_t)ET * 4;      // segment max       [N,4]
  float* DEN  = EMAX + (size_t)N * 4;      // segment denom     [N,4]

  const int gemmBlk  = ((N / 16) + 7) / 8;       // wave per 16-row strip
  const int edgeBlk  = (ET + 7) / 8;             // wave per edge
  const int nodeBlk  = (N + 7) / 8;              // wave per node
  auto fillBlk = [](size_t n) { return (int)((n + 255) / 256); };

  // ======================= Layer 0 (K=256, H=4, C=32) =======================
  k_gemm_bias<256,128><<<gemmBlk,256,0,stream>>>(xin, WL[0], BL[0], XL,  N);
  k_gemm_bias<256,128><<<gemmBlk,256,0,stream>>>(xin, WR[0], BR[0], XRA, N);
  k_gemm_bias<256,128><<<gemmBlk,256,0,stream>>>(xin, RW[0], RB[0], RES, N);
  k_fill<<<fillBlk((size_t)N*4),256,0,stream>>>(EMAX, FLT_MAX_NEG, (size_t)N*4);
  k_fill<<<fillBlk((size_t)N*4),256,0,stream>>>(DEN, 0.f, (size_t)N*4);
  k_edge_scores<4,32><<<edgeBlk,256,0,stream>>>(XL, XRA, ei, E, ET, ATT[0], ESC, EMAX);
  k_fill<<<fillBlk((size_t)N*128),256,0,stream>>>(XRA, 0.f, (size_t)N*128);
  k_edge_exp<4><<<fillBlk((size_t)ET*4),256,0,stream>>>(ei, E, ET, ESC, EMAX, DEN);
  k_edge_scatter<4,32><<<edgeBlk,256,0,stream>>>(XL, ei, E, ET, ESC, DEN, XRA);
  k_node_finish<<<nodeBlk,256,0,stream>>>(XRA, RES, BIAS[0], GAM[0], BET[0], F0, N);

  // ======================= Layer 1 (K=128, H=4, C=32) =======================
  k_gemm_bias<128,128><<<gemmBlk,256,0,stream>>>(F0, WL[1], BL[1], XL,  N);
  k_gemm_bias<128,128><<<gemmBlk,256,0,stream>>>(F0, WR[1], BR[1], XRA, N);
  k_gemm_bias<128,128><<<gemmBlk,256,0,stream>>>(F0, RW[1], RB[1], RES, N);
  k_fill<<<fillBlk((size_t)N*4),256,0,stream>>>(EMAX, FLT_MAX_NEG, (size_t)N*4);
  k_fill<<<fillBlk((size_t)N*4),256,0,stream>>>(DEN, 0.f, (size_t)N*4);
  k_edge_scores<4,32><<<edgeBlk,256,0,stream>>>(XL, XRA, ei, E, ET, ATT[1], ESC, EMAX);
  k_fill<<<fillBlk((size_t)N*128),256,0,stream>>>(XRA, 0.f, (size_t)N*128);
  k_edge_exp<4><<<fillBlk((size_t)ET*4),256,0,stream>>>(ei, E, ET, ESC, EMAX, DEN);
  k_edge_scatter<4,32><<<edgeBlk,256,0,stream>>>(XL, ei, E, ET, ESC, DEN, XRA);
  k_node_finish<<<nodeBlk,256,0,stream>>>(XRA, RES, BIAS[1], GAM[1], BET[1], F0, N);

  // ======================= Layer 2 (K=128, H=1, C=64) =======================
  k_gemm_bias<128,64><<<gemmBlk,256,0,stream>>>(F0, WL[2], BL[2], XL,  N);
  k_gemm_bias<128,64><<<gemmBlk,256,0,stream>>>(F0, WR[2], BR[2], XRA, N);
  k_fill<<<fillBlk((size_t)N),256,0,stream>>>(EMAX, FLT_MAX_NEG, (size_t)N);
  k_fill<<<fillBlk((size_t)N),256,0,stream>>>(DEN, 0.f, (size_t)N);
  k_edge_scores<1,64><<<edgeBlk,256,0,stream>>>(XL, XRA, ei, E, ET, ATT[2], ESC, EMAX);
  k_fill<<<fillBlk((size_t)N*64),256,0,stream>>>(XRA, 0.f, (size_t)N*64);
  k_edge_exp<1><<<fillBlk((size_t)ET),256,0,stream>>>(ei, E, ET, ESC, EMAX, DEN);
  k_edge_scatter<1,64><<<edgeBlk,256,0,stream>>>(XL, ei, E, ET, ESC, DEN, XRA);
  k_final_bias<<<fillBlk((size_t)N*64),256,0,stream>>>(XRA, BIAS[2], (float*)d_out, N);
}